// QuanvolutionHybrid_65481071409463
// MI455X (gfx1250) — compile-verified
//
#include <hip/hip_runtime.h>
#include <math.h>

// CDNA5 / gfx1250: wave32, WMMA f32 16x16x4 (full-precision matrix core path).
typedef __attribute__((ext_vector_type(2))) float v2f;
typedef __attribute__((ext_vector_type(8))) float v8f;

#define NBATCH 4096
#define NPATCH 196
#define NFEAT  784   // NPATCH * 4

// ---------------------------------------------------------------------------
// Kernel 1: quanvolution features.
// One wave handles a 16-patch tile (patches flattened over batch x patch).
// A-matrix (16x4 f32): row M = patch (lane%16), K = state-index chunk.
//   VGPR0 = K0 (lanes 0-15) / K2 (lanes 16-31); VGPR1 = K1 / K3.
// B-matrix (4x16 f32): U^T chunk, N = output state j (lane%16).
// D (16x16 f32): lane holds N = lane%16, rows M = r + 8*(lane/16) per VGPR r.
// probs -> Pauli-Z via 16-point signed WHT across the N (lane) dimension.
// ---------------------------------------------------------------------------
__global__ __launch_bounds__(256) void quanv_wmma_kernel(
    const float* __restrict__ x,
    const float* __restrict__ Ure,
    const float* __restrict__ Uim,
    float* __restrict__ feat)
{
  const int lane = threadIdx.x & 31;
  const int wave = threadIdx.x >> 5;
  const int tile = blockIdx.x * 8 + wave;      // 50176 tiles total
  const int m    = lane & 15;                  // A row (patch) / B col (state j)
  const int half = lane >> 4;                  // K sub-row selector

  // --- per-lane patch -> product state st[16] -------------------------------
  const int g  = tile * 16 + m;                // global patch id
  const int b  = g / NPATCH;
  const int pp = g - b * NPATCH;
  const int pr = pp / 14;
  const int pc = pp - pr * 14;

  const float* px = x + b * 784 + (pr * 2) * 28 + (pc * 2);
  float2 row0 = *(const float2*)(px);
  float2 row1 = *(const float2*)(px + 28);

  float s0 = __sinf(0.5f * row0.x), c0 = __cosf(0.5f * row0.x);
  float s1 = __sinf(0.5f * row0.y), c1 = __cosf(0.5f * row0.y);
  float s2 = __sinf(0.5f * row1.x), c2 = __cosf(0.5f * row1.x);
  float s3 = __sinf(0.5f * row1.y), c3 = __cosf(0.5f * row1.y);

  // st[i*8+j*4+k*2+l] = q0[i]*q1[j]*q2[k]*q3[l], wire0 = MSB
  float t0 = c0 * c1, t1 = c0 * s1, t2 = s0 * c1, t3 = s0 * s1;
  float u0 = c2 * c3, u1 = c2 * s3, u2 = s2 * c3, u3 = s2 * s3;
  float st[16];
  st[ 0] = t0*u0; st[ 1] = t0*u1; st[ 2] = t0*u2; st[ 3] = t0*u3;
  st[ 4] = t1*u0; st[ 5] = t1*u1; st[ 6] = t1*u2; st[ 7] = t1*u3;
  st[ 8] = t2*u0; st[ 9] = t2*u1; st[10] = t2*u2; st[11] = t2*u3;
  st[12] = t3*u0; st[13] = t3*u1; st[14] = t3*u2; st[15] = t3*u3;

  // --- st2 = st @ U^T, split into real/imag f32 WMMAs ----------------------
  v8f dre = {};
  v8f dim_ = {};
#pragma unroll
  for (int kk = 0; kk < 4; ++kk) {
    v2f a;
    a.x = half ? st[kk * 4 + 2] : st[kk * 4 + 0];
    a.y = half ? st[kk * 4 + 3] : st[kk * 4 + 1];
    const int col = kk * 4 + half * 2;           // global K for this lane-half
    v2f br, bi;
    br.x = Ure[m * 16 + col];  br.y = Ure[m * 16 + col + 1];
    bi.x = Uim[m * 16 + col];  bi.y = Uim[m * 16 + col + 1];
    dre  = __builtin_amdgcn_wmma_f32_16x16x4_f32(false, a, false, br,
                                                 (short)0, dre,  false, false);
    dim_ = __builtin_amdgcn_wmma_f32_16x16x4_f32(false, a, false, bi,
                                                 (short)0, dim_, false, false);
  }

  // --- probs -> Walsh-Hadamard over state index (lane dim) -----------------
#pragma unroll
  for (int r = 0; r < 8; ++r) {
    float p = dre[r] * dre[r] + dim_[r] * dim_[r];
#pragma unroll
    for (int msk = 1; msk < 16; msk <<= 1) {
      float y = __shfl_xor(p, msk, 32);
      p = (lane & msk) ? (y - p) : (p + y);
    }
    // lane j = m holds WHT coeff j; wires live at j = 8,4,2,1 (w = 0..3)
    if (m == 1 || m == 2 || m == 4 || m == 8) {
      const int w  = (m == 8) ? 0 : (m == 4) ? 1 : (m == 2) ? 2 : 3;
      const int gp = tile * 16 + r + half * 8;   // D-row patch id
      feat[gp * 4 + w] = p;
    }
  }
}

// ---------------------------------------------------------------------------
// Kernel 2: logits = feat @ W^T + b, then log-softmax. M=4096, K=784, N=10->16.
// One 16-sample tile per block; 4 waves split K (49 chunks each, with two
// interleaved accumulators to break the WMMA dependency chain), partials
// reduced through LDS, wave 0 runs the masked log-softmax epilogue.
// ---------------------------------------------------------------------------
__global__ __launch_bounds__(128) void head_wmma_kernel(
    const float* __restrict__ feat,
    const float* __restrict__ W,
    const float* __restrict__ bias,
    float* __restrict__ out)
{
  __shared__ float partial[4][32][8];

  const int lane  = threadIdx.x & 31;
  const int wave  = threadIdx.x >> 5;          // K-slice 0..3
  const int tile  = blockIdx.x;                // 256 tiles of 16 samples
  const int n     = lane & 15;                 // class index (padded to 16)
  const int half  = lane >> 4;

  const float* arow = feat + (size_t)(tile * 16 + n) * NFEAT; // A row M=n
  const float* wrow = W + (size_t)n * NFEAT;                  // B col N=n

  v8f acc0 = {};
  v8f acc1 = {};
  const int kk0 = wave * 49;
#pragma unroll 2
  for (int i = 0; i < 49; ++i) {
    const int k = (kk0 + i) * 4 + half * 2;
    float2 av = *(const float2*)(arow + k);
    v2f a;  a.x = av.x;  a.y = av.y;
    v2f bw; bw.x = 0.0f; bw.y = 0.0f;
    if (n < 10) {
      float2 wv = *(const float2*)(wrow + k);
      bw.x = wv.x; bw.y = wv.y;
    }
    if (i & 1)
      acc1 = __builtin_amdgcn_wmma_f32_16x16x4_f32(false, a, false, bw,
                                                   (short)0, acc1, false, false);
    else
      acc0 = __builtin_amdgcn_wmma_f32_16x16x4_f32(false, a, false, bw,
                                                   (short)0, acc0, false, false);
  }

#pragma unroll
  for (int r = 0; r < 8; ++r)
    partial[wave][lane][r] = acc0[r] + acc1[r];

  __syncthreads();
  if (wave != 0) return;

  const float bn = (n < 10) ? bias[n] : 0.0f;
#pragma unroll
  for (int r = 0; r < 8; ++r) {
    float v = partial[0][lane][r] + partial[1][lane][r] +
              partial[2][lane][r] + partial[3][lane][r] + bn;
    float vm = (n < 10) ? v : -INFINITY;
#pragma unroll
    for (int msk = 1; msk < 16; msk <<= 1)
      vm = fmaxf(vm, __shfl_xor(vm, msk, 32));
    float e = (n < 10) ? __expf(v - vm) : 0.0f;
#pragma unroll
    for (int msk = 1; msk < 16; msk <<= 1)
      e += __shfl_xor(e, msk, 32);
    const float lsm = (v - vm) - __logf(e);
    if (n < 10) {
      const int s = tile * 16 + r + half * 8;  // D-row sample id
      out[s * 10 + n] = lsm;
    }
  }
}

extern "C" void kernel_launch(void* const* d_in, const int* in_sizes, int n_in,
                              void* d_out, int out_size, void* d_ws, size_t ws_size,
                              hipStream_t stream) {
  const float* x    = (const float*)d_in[0];  // [4096,28,28]
  const float* Ure  = (const float*)d_in[1];  // [16,16]
  const float* Uim  = (const float*)d_in[2];  // [16,16]
  const float* W    = (const float*)d_in[3];  // [10,784]
  const float* bias = (const float*)d_in[4];  // [10]
  float* out  = (float*)d_out;                // [4096,10]
  float* feat = (float*)d_ws;                 // [4096,784] = 12.8 MB scratch

  // 4096*196 patches / 16 per tile = 50176 tiles, 8 waves (tiles) per block
  quanv_wmma_kernel<<<6272, 256, 0, stream>>>(x, Ure, Uim, feat);
  // 256 tiles, 1 tile per 4-wave block (K split 4 ways)
  head_wmma_kernel<<<256, 128, 0, stream>>>(feat, W, bias, out);
}